// CosineEmbeddingBCELossSumWithWeightedNegtive_74801150427885
// MI455X (gfx1250) — compile-verified
//
#include <hip/hip_runtime.h>
#include <math.h>

typedef __attribute__((ext_vector_type(2))) float v2f;
typedef __attribute__((ext_vector_type(8))) float v8f;

#define BN    4096      // batch (rows of A, rows of Bm)
#define DK    256       // feature dim (GEMM K)
#define TM    128       // C tile rows per workgroup
#define TN    128       // C tile cols per workgroup
#define KB    32        // K chunk staged in LDS
#define KP    (KB / 2)  // K-pairs per chunk
#define LSTR2 (TM + 2)  // stage stride in v2f units (130)
#define CSTR  TN        // epilogue tile stride (floats); diag step 129 -> bank step 1

// ---------------------------------------------------------------------------
// Kernel 1: row L2 norms of A and B into workspace; zero the loss scalar.
// ---------------------------------------------------------------------------
__global__ __launch_bounds__(256)
void norms_zero_kernel(const float* __restrict__ A, const float* __restrict__ Bm,
                       float* __restrict__ na, float* __restrict__ nb,
                       float* __restrict__ loss) {
  const int row = blockIdx.x;
  const int t = threadIdx.x;
  if (row == 0 && t == 0) loss[0] = 0.0f;   // deterministic re-init each call

  float va = A[row * DK + t];
  float vb = Bm[row * DK + t];
  float sa = va * va;
  float sb = vb * vb;
#pragma unroll
  for (int off = 16; off > 0; off >>= 1) {   // wave32 reduce
    sa += __shfl_down(sa, off, 32);
    sb += __shfl_down(sb, off, 32);
  }
  __shared__ float wsa[8], wsb[8];
  if ((t & 31) == 0) { wsa[t >> 5] = sa; wsb[t >> 5] = sb; }
  __syncthreads();
  if (t == 0) {
    float ta = 0.f, tb = 0.f;
#pragma unroll
    for (int w = 0; w < 8; w++) { ta += wsa[w]; tb += wsb[w]; }
    na[row] = sqrtf(ta);
    nb[row] = sqrtf(tb);
  }
}

// ---------------------------------------------------------------------------
// Kernel 2: 128x128 tile of C = A * Bm^T via V_WMMA_F32_16X16X4_F32.
// K-pair-interleaved LDS staging: each fragment operand is one aligned
// ds_load_b64 directly into an even VGPR pair (no mov repacking).
// Register prefetch pipeline overlaps next-chunk HBM loads with WMMA.
// Epilogue: LDS diagonal walk -> cosine scale -> coalesced scatter to
// out[((j-k)&4095)*4096 + k], fused weighted-BCE sum.
// ---------------------------------------------------------------------------
__global__ __launch_bounds__(256)
void gemm_cos_bce_kernel(const float* __restrict__ A, const float* __restrict__ Bm,
                         const float* __restrict__ labels,
                         const float* __restrict__ na, const float* __restrict__ nb,
                         float* __restrict__ out, float* __restrict__ loss) {
  extern __shared__ float smem[];
  v2f* As2 = (v2f*)smem;             // [KP][LSTR2]  A stage, K-pair major
  v2f* Bs2 = As2 + KP * LSTR2;       // [KP][LSTR2]  B stage
  float* Csl = smem;                 // [TM][CSTR]   epilogue tile (reuses stage)

  const int i0 = blockIdx.x * TM;    // C row base (problem index k)
  const int j0 = blockIdx.y * TN;    // C col base (problem index j)
  const int tid = threadIdx.x;
  const int lane = tid & 31;
  const int wid = tid >> 5;          // 8 waves: 2 (m) x 4 (n)
  const int wave_m = (wid & 1) * 64; // 4 fragments of 16 in m
  const int wave_n = (wid >> 1) * 32;// 2 fragments of 16 in n
  const int lhalf = lane >> 4;       // half-wave: K pairs {ks,ks+1} vs {ks+2,ks+3}
  const int l16 = lane & 15;

  const int dp = tid & 15;           // k-pair index this thread stages
  const int mb = tid >> 4;           // row base this thread stages (step 16/iter)

  v8f acc[4][2] = {};
  v2f rA[8], rB[8];

  // Prefetch chunk 0 (coalesced v2f loads: 128B per 16 lanes).
#pragma unroll
  for (int p = 0; p < 8; p++) {
    int m = mb + 16 * p;
    rA[p] = *(const v2f*)(A  + (size_t)(i0 + m) * DK + 2 * dp);
    rB[p] = *(const v2f*)(Bm + (size_t)(j0 + m) * DK + 2 * dp);
  }

  for (int kc = 0; kc < DK; kc += KB) {
    __syncthreads();
#pragma unroll
    for (int p = 0; p < 8; p++) {    // regs -> LDS (disjoint bank pairs)
      int m = mb + 16 * p;
      As2[dp * LSTR2 + m] = rA[p];
      Bs2[dp * LSTR2 + m] = rB[p];
    }
    __syncthreads();

    if (kc + KB < DK) {              // issue next-chunk HBM loads now;
#pragma unroll                       // the 64 WMMAs below hide the latency
      for (int p = 0; p < 8; p++) {
        int m = mb + 16 * p;
        rA[p] = *(const v2f*)(A  + (size_t)(i0 + m) * DK + (kc + KB) + 2 * dp);
        rB[p] = *(const v2f*)(Bm + (size_t)(j0 + m) * DK + (kc + KB) + 2 * dp);
      }
    }

#pragma unroll
    for (int ks = 0; ks < KB; ks += 4) {
      // ISA f32 A 16x4 layout: lanes0-15 hold K={ks,ks+1}, lanes16-31 K={ks+2,ks+3};
      // stored as one v2f -> single aligned ds_load_b64 per fragment.
      const int kp = (ks >> 1) + lhalf;
      v2f af[4], bf[2];
#pragma unroll
      for (int fm = 0; fm < 4; fm++)
        af[fm] = As2[kp * LSTR2 + wave_m + fm * 16 + l16];
#pragma unroll
      for (int fn = 0; fn < 2; fn++)
        bf[fn] = Bs2[kp * LSTR2 + wave_n + fn * 16 + l16];
#pragma unroll
      for (int fm = 0; fm < 4; fm++)
#pragma unroll
        for (int fn = 0; fn < 2; fn++)
          acc[fm][fn] = __builtin_amdgcn_wmma_f32_16x16x4_f32(
              false, af[fm], false, bf[fn], (short)0, acc[fm][fn], false, false);
    }
  }

  __syncthreads();
  // C fragment layout: lane -> N = lane%16, VGPR r + 8*half -> M.
#pragma unroll
  for (int fm = 0; fm < 4; fm++)
#pragma unroll
    for (int fn = 0; fn < 2; fn++)
#pragma unroll
      for (int r = 0; r < 8; r++) {
        int m = wave_m + fm * 16 + (lhalf << 3) + r;
        int n = wave_n + fn * 16 + l16;
        Csl[m * CSTR + n] = acc[fm][fn][r];
      }
  __syncthreads();

  // Wrapped-diagonal walk: u = diag shift, dk = row-in-tile, dj = (dk+u)&127.
  // Consecutive lanes -> consecutive global addresses (coalesced runs);
  // LDS read stride 129 floats -> bank step 1 (conflict-free).
  float lsum = 0.0f;
  const int sbase = (j0 - i0) & (BN - 1);
  const int v = tid & 127;           // dk
  const int uoff = tid >> 7;         // two diagonals per iteration
#pragma unroll 1
  for (int ub = 0; ub < TM; ub += 2) {
    int u = ub + uoff;
    int dk = v;
    int dj = (dk + u) & (TM - 1);
    float dot = Csl[dk * CSTR + dj];
    int i = i0 + dk;
    int j = j0 + dj;
    float denom = fmaxf(na[i] * nb[j], 1e-8f);
    float p = (dot / denom + 1.0f) * 0.5f;
    int s = (sbase + dj - dk) & (BN - 1);
    out[(size_t)s * BN + i] = p;
    // BCE(sum): s==0 block -> weight 1, label = labels[i]; else weight .5, label 0.
    if (s == 0) {
      float y = labels[i];
      lsum -= y * fmaxf(logf(p), -100.0f) + (1.0f - y) * fmaxf(log1pf(-p), -100.0f);
    } else {
      lsum -= 0.5f * fmaxf(log1pf(-p), -100.0f);
    }
  }
#pragma unroll
  for (int off = 16; off > 0; off >>= 1) lsum += __shfl_down(lsum, off, 32);
  if ((tid & 31) == 0) atomicAdd(loss, lsum);
}

// ---------------------------------------------------------------------------
extern "C" void kernel_launch(void* const* d_in, const int* in_sizes, int n_in,
                              void* d_out, int out_size, void* d_ws, size_t ws_size,
                              hipStream_t stream) {
  const float* A      = (const float*)d_in[0];   // feature_a [4096,256] f32
  const float* Bm     = (const float*)d_in[1];   // feature_b [4096,256] f32
  const float* labels = (const float*)d_in[2];   // [4096] f32

  float* out  = (float*)d_out;                   // logits [4096*4096] f32
  float* loss = out + (size_t)BN * BN;           // scalar loss appended
  float* na = (float*)d_ws;                      // [4096]
  float* nb = na + BN;                           // [4096]

  norms_zero_kernel<<<BN, 256, 0, stream>>>(A, Bm, na, nb, loss);

  dim3 grid(BN / TM, BN / TN);                   // 32 x 32 tiles
  size_t shmem = (size_t)TM * CSTR * sizeof(float); // 64 KB (covers 33.3 KB stage)
  gemm_cos_bce_kernel<<<grid, 256, shmem, stream>>>(A, Bm, labels, na, nb, out, loss);
}